// InvLinear_77756087926982
// MI455X (gfx1250) — compile-verified
//
#include <hip/hip_runtime.h>

// InvLinear fused kernel for MI455X (gfx1250), wave32 + WMMA f32 16x16x4.
//
// out = (merged*(63/64) - Xlast/64) @ Wl^T + bl
// merged[n,d] = sum_i lrelu(X@Wg^T + bg)[n,i] * X[n, i*64+d]

typedef float v2f __attribute__((ext_vector_type(2)));
typedef float v8f __attribute__((ext_vector_type(8)));

#define IN_F        4096
#define OUT_F       64
#define DIM         64
#define ROWS_PER_WAVE   16
#define WAVES_PER_BLOCK 8
#define ROWS_PER_BLOCK  (ROWS_PER_WAVE * WAVES_PER_BLOCK)   // 128
#define LDS_STRIDE  68   // 64 + 4 pad: conflict-free ds reads/writes everywhere

__global__ __launch_bounds__(256)
void invlinear_fused_kernel(const float* __restrict__ X,
                            const float* __restrict__ Wg,
                            const float* __restrict__ bg,
                            const float* __restrict__ Wl,
                            const float* __restrict__ bl,
                            float* __restrict__ out) {
    __shared__ float smem[WAVES_PER_BLOCK * ROWS_PER_WAVE * LDS_STRIDE];

    const int tid  = threadIdx.x;
    const int wave = tid >> 5;
    const int lane = tid & 31;
    const int half = lane >> 4;        // 0: lanes 0-15, 1: lanes 16-31
    const int l15  = lane & 15;
    const int koff = half * 2;         // f32 A/B frag K offset per lane-half

    const int rowBase = blockIdx.x * ROWS_PER_BLOCK + wave * ROWS_PER_WAVE;
    float* wls = smem + wave * ROWS_PER_WAVE * LDS_STRIDE;   // this wave's 16x68 tile

    // ---------------- Pass 1: alpha = leaky_relu(X @ Wg^T + bg) ----------------
    // A (16x4 f32): lane l15 = row M, v2f = K {koff, koff+1} within 4-wide step.
    // B (4x16 f32): lane l15 = col N, v2f = K {koff, koff+1}. B[k][n] = Wg[n][k].
    const float* pA  = X  + (size_t)(rowBase + l15) * IN_F + koff;
    const float* pB0 = Wg + (size_t)(l15 +  0) * IN_F + koff;
    const float* pB1 = Wg + (size_t)(l15 + 16) * IN_F + koff;
    const float* pB2 = Wg + (size_t)(l15 + 32) * IN_F + koff;
    const float* pB3 = Wg + (size_t)(l15 + 48) * IN_F + koff;

    v8f acc0 = {}, acc1 = {}, acc2 = {}, acc3 = {};

#pragma unroll 8
    for (int kb = 0; kb < IN_F; kb += 4) {
        v2f a  = *(const v2f*)(pA  + kb);
        v2f b0 = *(const v2f*)(pB0 + kb);
        v2f b1 = *(const v2f*)(pB1 + kb);
        v2f b2 = *(const v2f*)(pB2 + kb);
        v2f b3 = *(const v2f*)(pB3 + kb);
        acc0 = __builtin_amdgcn_wmma_f32_16x16x4_f32(false, a, false, b0, (short)0, acc0, false, false);
        acc1 = __builtin_amdgcn_wmma_f32_16x16x4_f32(false, a, false, b1, (short)0, acc1, false, false);
        acc2 = __builtin_amdgcn_wmma_f32_16x16x4_f32(false, a, false, b2, (short)0, acc2, false, false);
        acc3 = __builtin_amdgcn_wmma_f32_16x16x4_f32(false, a, false, b3, (short)0, acc3, false, false);
    }

    // Bias + LeakyReLU(0.1), park alpha (16x64) in LDS.
    // C/D layout: vgpr e, lanes 0-15 -> M=e, lanes 16-31 -> M=e+8; N = l15 + 16*nt.
    {
        const float bg0 = bg[l15], bg1 = bg[l15 + 16], bg2 = bg[l15 + 32], bg3 = bg[l15 + 48];
        const int rb = half * 8;
#pragma unroll
        for (int e = 0; e < 8; ++e) {
            float* dst = wls + (rb + e) * LDS_STRIDE + l15;
            float v0 = acc0[e] + bg0; v0 = v0 > 0.f ? v0 : 0.1f * v0;
            float v1 = acc1[e] + bg1; v1 = v1 > 0.f ? v1 : 0.1f * v1;
            float v2 = acc2[e] + bg2; v2 = v2 > 0.f ? v2 : 0.1f * v2;
            float v3 = acc3[e] + bg3; v3 = v3 > 0.f ? v3 : 0.1f * v3;
            dst[0]  = v0;
            dst[16] = v1;
            dst[32] = v2;
            dst[48] = v3;
        }
    }

    // ---------------- Pass 2: merged + fix-up -> pre, in-place over alpha -------
    // merged[r,d] = sum_i alpha[r,i] * X[r, 64*i + d]; lane owns d = {2*lane, 2*lane+1}.
    // X reload hits L2 (tile just streamed in pass 1); mark non-temporal (last use).
    {
        const int d2 = lane * 2;
#pragma unroll 1
        for (int r = 0; r < ROWS_PER_WAVE; ++r) {
            const float* xr = X + (size_t)(rowBase + r) * IN_F + d2;
            float m0 = 0.f, m1 = 0.f;
            v2f xi = {};
#pragma unroll 8
            for (int i = 0; i < OUT_F; ++i) {
                const float a = wls[r * LDS_STRIDE + i];           // LDS broadcast
                xi = __builtin_nontemporal_load((const v2f*)(xr + i * DIM));
                m0 += a * xi[0];
                m1 += a * xi[1];
            }
            // xi now holds Xlast = X[r, 63*64 + d]; pre = merged*(63/64) - Xlast/64
            v2f pre;
            pre[0] = m0 * (63.f / 64.f) - xi[0] * (1.f / 64.f);
            pre[1] = m1 * (63.f / 64.f) - xi[1] * (1.f / 64.f);
            *(v2f*)(wls + r * LDS_STRIDE + d2) = pre;              // alpha row r is dead
        }
    }

    // ---------------- Pass 3: out = pre @ Wl^T + bl (WMMA, K=64) ----------------
    {
        const float* qW0 = Wl + (size_t)(l15 +  0) * DIM + koff;
        const float* qW1 = Wl + (size_t)(l15 + 16) * DIM + koff;
        const float* qW2 = Wl + (size_t)(l15 + 32) * DIM + koff;
        const float* qW3 = Wl + (size_t)(l15 + 48) * DIM + koff;
        const float* aLds = wls + l15 * LDS_STRIDE + koff;         // stride-68 pad: no bank conflicts

        v8f o0 = {}, o1 = {}, o2 = {}, o3 = {};
#pragma unroll
        for (int kb = 0; kb < DIM; kb += 4) {
            v2f a  = *(const v2f*)(aLds + kb);
            v2f b0 = *(const v2f*)(qW0 + kb);
            v2f b1 = *(const v2f*)(qW1 + kb);
            v2f b2 = *(const v2f*)(qW2 + kb);
            v2f b3 = *(const v2f*)(qW3 + kb);
            o0 = __builtin_amdgcn_wmma_f32_16x16x4_f32(false, a, false, b0, (short)0, o0, false, false);
            o1 = __builtin_amdgcn_wmma_f32_16x16x4_f32(false, a, false, b1, (short)0, o1, false, false);
            o2 = __builtin_amdgcn_wmma_f32_16x16x4_f32(false, a, false, b2, (short)0, o2, false, false);
            o3 = __builtin_amdgcn_wmma_f32_16x16x4_f32(false, a, false, b3, (short)0, o3, false, false);
        }

        const float bl0 = bl[l15], bl1 = bl[l15 + 16], bl2 = bl[l15 + 32], bl3 = bl[l15 + 48];
        float* po = out + (size_t)(rowBase + half * 8) * OUT_F + l15;
#pragma unroll
        for (int e = 0; e < 8; ++e) {
            po[(size_t)e * OUT_F +  0] = o0[e] + bl0;
            po[(size_t)e * OUT_F + 16] = o1[e] + bl1;
            po[(size_t)e * OUT_F + 32] = o2[e] + bl2;
            po[(size_t)e * OUT_F + 48] = o3[e] + bl3;
        }
    }
}

extern "C" void kernel_launch(void* const* d_in, const int* in_sizes, int n_in,
                              void* d_out, int out_size, void* d_ws, size_t ws_size,
                              hipStream_t stream) {
    (void)in_sizes; (void)n_in; (void)out_size; (void)d_ws; (void)ws_size;
    const float* X  = (const float*)d_in[0];
    const float* Wg = (const float*)d_in[1];
    const float* bg = (const float*)d_in[2];
    const float* Wl = (const float*)d_in[3];
    const float* bl = (const float*)d_in[4];
    float* out = (float*)d_out;

    const int N = 65536;
    dim3 grid(N / ROWS_PER_BLOCK);   // 512 blocks
    dim3 block(256);                 // 8 waves of 32
    invlinear_fused_kernel<<<grid, block, 0, stream>>>(X, Wg, bg, Wl, bl, out);
}